// TemporalGNNEncoder_28355374088213
// MI455X (gfx1250) — compile-verified
//
#include <hip/hip_runtime.h>
#include <math.h>

typedef __attribute__((ext_vector_type(2))) float v2f;
typedef __attribute__((ext_vector_type(8))) float v8f;

// ---------------- small utility kernels ----------------

__global__ void k_fill(float* __restrict__ p, int n, float v) {
  int t = blockIdx.x * blockDim.x + threadIdx.x;
  if (t < n) p[t] = v;
}

__global__ void k_deg(const int* __restrict__ dst, float* __restrict__ deg, int e) {
  int t = blockIdx.x * blockDim.x + threadIdx.x;
  if (t < e) atomicAdd(&deg[dst[t]], 1.0f);
}

__global__ void k_rsqrt(float* __restrict__ p, int n) {
  int t = blockIdx.x * blockDim.x + threadIdx.x;
  if (t < n) {
    float d = p[t];
    p[t] = (d > 0.0f) ? rsqrtf(d) : 0.0f;
  }
}

__global__ void k_norm(const int* __restrict__ src, const int* __restrict__ dst,
                       const float* __restrict__ dinv, float* __restrict__ norm, int e) {
  int t = blockIdx.x * blockDim.x + threadIdx.x;
  if (t < e) norm[t] = dinv[src[t]] * dinv[dst[t]];
}

__global__ void k_relu(float* __restrict__ p, size_t n) {
  size_t t = (size_t)blockIdx.x * blockDim.x + threadIdx.x;
  if (t < n) p[t] = fmaxf(p[t], 0.0f);
}

__global__ void k_clamp10(float* __restrict__ p, size_t n) {
  size_t t = (size_t)blockIdx.x * blockDim.x + threadIdx.x;
  if (t < n) p[t] = fminf(p[t], 10.0f);
}

__global__ void k_bcast_add(float* __restrict__ h, const float* __restrict__ ht, size_t n) {
  size_t t = (size_t)blockIdx.x * blockDim.x + threadIdx.x;
  if (t < n) h[t] += ht[t & 127];   // H == 128
}

// ---------------- WMMA f32 GEMM: Y[N,KOUT] = X[N,KIN] @ W[KIN,KOUT] ----------------
// One block per 16-row strip. A strip staged in LDS (shared by all waves).
// One wave32 per 16x16 output tile, using V_WMMA_F32_16X16X4_F32, K stepped by 4.
template <int KIN, int KOUT>
__global__ void gemm_wmma_f32(const float* __restrict__ X, const float* __restrict__ W,
                              float* __restrict__ Y, int nrows) {
  __shared__ float As[16 * KIN];
  const int nthreads = 32 * (KOUT / 16);
  const int wave = threadIdx.x >> 5;
  const int lane = threadIdx.x & 31;
  const int m0 = blockIdx.x * 16;

  // cooperative load of the 16 x KIN strip of X into LDS
  for (int i = threadIdx.x; i < 16 * KIN; i += nthreads) {
    int r = i / KIN, c = i % KIN;
    As[i] = (m0 + r < nrows) ? X[(size_t)(m0 + r) * KIN + c] : 0.0f;
  }
  __syncthreads();

  const int n0 = wave * 16;
  const int l15 = lane & 15;   // A: row M; B/D: col N
  const int hi = lane >> 4;    // K-pair / row-half select

  v8f acc = {};
#pragma unroll
  for (int k = 0; k < KIN; k += 4) {
    // A 16x4 f32 layout: lanes 0-15 hold K={0,1}, lanes 16-31 hold K={2,3}
    v2f a;
    a.x = As[l15 * KIN + k + 2 * hi + 0];
    a.y = As[l15 * KIN + k + 2 * hi + 1];
    // B 4x16 f32 layout (mirrored): VGPR0 = rows K=0/K=2, VGPR1 = rows K=1/K=3
    v2f b;
    b.x = W[(size_t)(k + 2 * hi + 0) * KOUT + n0 + l15];
    b.y = W[(size_t)(k + 2 * hi + 1) * KOUT + n0 + l15];
    acc = __builtin_amdgcn_wmma_f32_16x16x4_f32(false, a, false, b, (short)0, acc,
                                                false, false);
  }

  // D 16x16 f32: VGPR i -> row M = i (lanes 0-15) / M = 8+i (lanes 16-31)
#pragma unroll
  for (int i = 0; i < 8; ++i) {
    int row = m0 + i + 8 * hi;
    if (row < nrows) Y[(size_t)row * KOUT + n0 + l15] = acc[i];
  }
}

// ---------------- GCN aggregation ----------------
// out[i,:] = bias + h[i,:]*dinv[i]^2  (self-loop term; also initializes out)
template <int F>
__global__ void agg_init(const float* __restrict__ h, const float* __restrict__ dinv,
                         const float* __restrict__ bias, float* __restrict__ out, int n) {
  size_t t = (size_t)blockIdx.x * blockDim.x + threadIdx.x;
  size_t total = (size_t)n * F;
  if (t >= total) return;
  int i = (int)(t / F);
  int f = (int)(t & (F - 1));
  float d = dinv[i];
  out[t] = bias[f] + h[t] * d * d;
}

// out[dst,:] += h[src,:] * norm[e]  -- float4 per thread, atomic scatter
template <int F>
__global__ void agg_edges(const float* __restrict__ h, const int* __restrict__ src,
                          const int* __restrict__ dst, const float* __restrict__ norm,
                          float* __restrict__ out, int nedges) {
  const int C = F / 4;
  size_t t = (size_t)blockIdx.x * blockDim.x + threadIdx.x;
  size_t total = (size_t)nedges * C;
  if (t >= total) return;
  int e = (int)(t / C);
  int c = (int)(t % C);
  int s = src[e], d = dst[e];
  float nm = norm[e];
  const float4 v = *reinterpret_cast<const float4*>(h + (size_t)s * F + c * 4);
  float* o = out + (size_t)d * F + c * 4;
  atomicAdd(o + 0, v.x * nm);
  atomicAdd(o + 1, v.y * nm);
  atomicAdd(o + 2, v.z * nm);
  atomicAdd(o + 3, v.w * nm);
}

// ---------------- mean pool + LSTM step ----------------

__global__ void k_pool(const float* __restrict__ h, float* __restrict__ sums, int n) {
  // blockDim = 128 (one feature per thread), each block covers 256 rows
  int f = threadIdx.x;
  int r0 = blockIdx.x * 256;
  int r1 = min(r0 + 256, n);
  float s = 0.0f;
  for (int r = r0; r < r1; ++r) s += h[(size_t)r * 128 + f];
  atomicAdd(&sums[f], s);
}

__global__ void k_lstm(const float* __restrict__ sums, const float* __restrict__ Wih,
                       const float* __restrict__ bih, const float* __restrict__ bhh,
                       float* __restrict__ ht, float invN) {
  __shared__ float ge[128];
  __shared__ float gates[512];
  int t = threadIdx.x;  // 512 threads
  if (t < 128) ge[t] = sums[t] * invN;
  __syncthreads();
  float acc = bih[t] + bhh[t];
#pragma unroll 8
  for (int k = 0; k < 128; ++k) acc += ge[k] * Wih[(size_t)t * 128 + k];
  gates[t] = acc;
  __syncthreads();
  if (t < 128) {
    float gi = gates[t];           // i
    float gg = gates[256 + t];     // g   (f gate unused: c0 = 0)
    float go = gates[384 + t];     // o
    float si = 1.0f / (1.0f + expf(-gi));
    float so = 1.0f / (1.0f + expf(-go));
    float c = si * tanhf(gg);
    ht[t] = so * tanhf(c);
  }
}

// ---------------- launcher ----------------

static inline unsigned cdiv(size_t a, size_t b) { return (unsigned)((a + b - 1) / b); }

extern "C" void kernel_launch(void* const* d_in, const int* in_sizes, int n_in,
                              void* d_out, int out_size, void* d_ws, size_t ws_size,
                              hipStream_t stream) {
  const int IN = 128, H = 128, OUT = 64;
  const int N = in_sizes[0] / IN;
  const int E = in_sizes[1] / 2;

  const float* x   = (const float*)d_in[0];
  const int*   ei  = (const int*)d_in[1];
  const int*   src = ei;
  const int*   dst = ei + E;
  const float* W1  = (const float*)d_in[2];
  const float* b1  = (const float*)d_in[3];
  const float* W2  = (const float*)d_in[4];
  const float* b2  = (const float*)d_in[5];
  const float* Wih = (const float*)d_in[6];
  // d_in[7] = W_hh (unused: h0 = 0)
  const float* bih = (const float*)d_in[8];
  const float* bhh = (const float*)d_in[9];
  const float* Wmu = (const float*)d_in[10];
  const float* bmu = (const float*)d_in[11];
  const float* Wls = (const float*)d_in[12];
  const float* bls = (const float*)d_in[13];

  float* out_mu = (float*)d_out;
  float* out_ls = out_mu + (size_t)N * OUT;

  // workspace layout
  float* dinv = (float*)d_ws;
  float* norm = dinv + N;
  float* bufA = norm + E;                    // N*H  (GEMM output)
  float* bufB = bufA + (size_t)N * H;        // N*H  (conv output)
  float* bufC = bufB + (size_t)N * H;        // N*OUT (head GEMM output)
  float* sums = bufC + (size_t)N * OUT;      // 128
  float* ht   = sums + 128;                  // 128

  const size_t NH = (size_t)N * H;
  const size_t NO = (size_t)N * OUT;
  const unsigned strips = cdiv((size_t)N, 16);

  // ---- degree / normalization (deg >= 1 due to self loops) ----
  k_fill<<<cdiv(N, 256), 256, 0, stream>>>(dinv, N, 1.0f);
  k_deg<<<cdiv(E, 256), 256, 0, stream>>>(dst, dinv, E);
  k_rsqrt<<<cdiv(N, 256), 256, 0, stream>>>(dinv, N);
  k_norm<<<cdiv(E, 256), 256, 0, stream>>>(src, dst, dinv, norm, E);

  // ---- GCN conv 1: relu(agg(x@W1) + b1) ----
  gemm_wmma_f32<128, 128><<<strips, 256, 0, stream>>>(x, W1, bufA, N);
  agg_init<128><<<cdiv(NH, 256), 256, 0, stream>>>(bufA, dinv, b1, bufB, N);
  agg_edges<128><<<cdiv((size_t)E * 32, 256), 256, 0, stream>>>(bufA, src, dst, norm, bufB, E);
  k_relu<<<cdiv(NH, 256), 256, 0, stream>>>(bufB, NH);

  // ---- GCN conv 2 ----
  gemm_wmma_f32<128, 128><<<strips, 256, 0, stream>>>(bufB, W2, bufA, N);
  agg_init<128><<<cdiv(NH, 256), 256, 0, stream>>>(bufA, dinv, b2, bufB, N);
  agg_edges<128><<<cdiv((size_t)E * 32, 256), 256, 0, stream>>>(bufA, src, dst, norm, bufB, E);
  k_relu<<<cdiv(NH, 256), 256, 0, stream>>>(bufB, NH);

  // ---- mean pool -> LSTM single step -> broadcast add ----
  k_fill<<<1, 128, 0, stream>>>(sums, 128, 0.0f);
  k_pool<<<cdiv(N, 256), 128, 0, stream>>>(bufB, sums, N);
  k_lstm<<<1, 512, 0, stream>>>(sums, Wih, bih, bhh, ht, 1.0f / (float)N);
  k_bcast_add<<<cdiv(NH, 256), 256, 0, stream>>>(bufB, ht, NH);

  // ---- mu head ----
  gemm_wmma_f32<128, 64><<<strips, 128, 0, stream>>>(bufB, Wmu, bufC, N);
  agg_init<64><<<cdiv(NO, 256), 256, 0, stream>>>(bufC, dinv, bmu, out_mu, N);
  agg_edges<64><<<cdiv((size_t)E * 16, 256), 256, 0, stream>>>(bufC, src, dst, norm, out_mu, E);

  // ---- logstd head (clamped at 10) ----
  gemm_wmma_f32<128, 64><<<strips, 128, 0, stream>>>(bufB, Wls, bufC, N);
  agg_init<64><<<cdiv(NO, 256), 256, 0, stream>>>(bufC, dinv, bls, out_ls, N);
  agg_edges<64><<<cdiv((size_t)E * 16, 256), 256, 0, stream>>>(bufC, src, dst, norm, out_ls, E);
  k_clamp10<<<cdiv(NO, 256), 256, 0, stream>>>(out_ls, NO);
}